// GCN_2061584302577
// MI455X (gfx1250) — compile-verified
//
#include <hip/hip_runtime.h>
#include <math.h>

// GCN + SAGPool pipeline for gfx1250 (MI455X).
// Dense transforms: V_WMMA_F32_16X16X4_F32, guard-free (padded shapes),
// A from global (b64 loads), B staged in LDS, fully unrolled per layer shape.
// Edge aggregation: coalesced per-(edge,feature) global_atomic_add_f32.
// Top-k: bitonic sort with stable (index) tie-break = jnp.argsort(-score).

typedef __attribute__((ext_vector_type(2))) float v2f;
typedef __attribute__((ext_vector_type(8))) float v8f;

#define TPB 256
static inline int cdiv(int a, int b) { return (a + b - 1) / b; }

// ---------------- dense transform: H = X @ W  (f32 WMMA, guard-free) -------
// X: [>=nTiles*16, FIN] (pad rows zero-filled by caller), W: [FIN, FOUT].
template <int FIN, int FOUT>
__global__ void k_wmma_gemm(const float* __restrict__ X, const float* __restrict__ W,
                            float* __restrict__ H, int nTilesRow) {
  __shared__ float Ws[FIN * FOUT];
  for (int t = threadIdx.x; t < FIN * FOUT; t += blockDim.x) Ws[t] = W[t];
  __syncthreads();

  int wave = (int)((blockIdx.x * blockDim.x + threadIdx.x) >> 5);
  int lane = (int)(threadIdx.x & 31);
  constexpr int NCT = FOUT / 16;
  int rowTile = wave / NCT;
  int colTile = wave - rowTile * NCT;
  if (rowTile >= nTilesRow) return;      // wave-uniform: EXEC all-1s at WMMA
  int row0 = rowTile << 4;
  int n0 = colTile << 4;
  int m  = lane & 15;
  int ks = lane >> 4;

  const float* xr = X + (size_t)(row0 + m) * FIN;   // 8B-aligned (FIN even)
  const float* wc = Ws + n0 + m;
  v8f acc = {0.f, 0.f, 0.f, 0.f, 0.f, 0.f, 0.f, 0.f};
#pragma unroll
  for (int k0 = 0; k0 < FIN; k0 += 4) {
    int ka = k0 + (ks << 1);
    v2f a = *(const v2f*)(xr + ka);      // A[m][ka], A[m][ka+1]
    v2f b;
    b.x = wc[ka * FOUT];                 // B[ka][n0+m]
    b.y = wc[(ka + 1) * FOUT];           // B[ka+1][n0+m]
    acc = __builtin_amdgcn_wmma_f32_16x16x4_f32(false, a, false, b,
                                                (short)0, acc, false, false);
  }
#pragma unroll
  for (int q = 0; q < 8; ++q)
    H[(size_t)(row0 + q + (ks << 3)) * FOUT + n0 + m] = acc[q];
}

static void launch_gemm(int Fin, int Fout, const float* X, const float* W,
                        float* H, int nTiles, hipStream_t s) {
  int waves = nTiles * (Fout / 16);
  int blocks = cdiv(waves * 32, TPB);
  if      (Fin == 4  && Fout == 16) k_wmma_gemm<4, 16><<<blocks, TPB, 0, s>>>(X, W, H, nTiles);
  else if (Fin == 16 && Fout == 32) k_wmma_gemm<16, 32><<<blocks, TPB, 0, s>>>(X, W, H, nTiles);
  else if (Fin == 32 && Fout == 64) k_wmma_gemm<32, 64><<<blocks, TPB, 0, s>>>(X, W, H, nTiles);
  else if (Fin == 64 && Fout == 32) k_wmma_gemm<64, 32><<<blocks, TPB, 0, s>>>(X, W, H, nTiles);
  else if (Fin == 32 && Fout == 32) k_wmma_gemm<32, 32><<<blocks, TPB, 0, s>>>(X, W, H, nTiles);
}

// ---------------- small utility kernels ----------------
__global__ void k_fill_f32(float* p, float v, int n) {
  int i = blockIdx.x * blockDim.x + threadIdx.x;
  if (i < n) p[i] = v;
}
__global__ void k_fill_i32(int* p, int v, int n) {
  int i = blockIdx.x * blockDim.x + threadIdx.x;
  if (i < n) p[i] = v;
}
// dst[t] = t < nSrc ? src[t] : 0   (append zero rows, same row width)
__global__ void k_copy_pad(float* dst, const float* __restrict__ src, int nSrc, int nTot) {
  int t = blockIdx.x * blockDim.x + threadIdx.x;
  if (t < nTot) dst[t] = (t < nSrc) ? src[t] : 0.f;
}
// widen rows: dst[r][c] = c < cs ? src[r][c] : 0
__global__ void k_pad_cols(float* dst, const float* __restrict__ src,
                           int rows, int cs, int cd) {
  int t = blockIdx.x * blockDim.x + threadIdx.x;
  if (t >= rows * cd) return;
  int r = t / cd, c = t - r * cd;
  dst[t] = (c < cs) ? src[r * cs + c] : 0.f;
}

// deg[c] += w[e]  (deg pre-filled with 1.0 for the self loop)
__global__ void k_deg_scatter(float* deg, const int* __restrict__ col,
                              const float* __restrict__ w, int E) {
  int e = blockIdx.x * blockDim.x + threadIdx.x;
  if (e >= E) return;
  float we = w[e];
  if (we != 0.f) atomicAdd(&deg[col[e]], we);
}
__global__ void k_rsqrt_inplace(float* deg, int n) {
  int i = blockIdx.x * blockDim.x + threadIdx.x;
  if (i < n) {
    float d = deg[i];
    deg[i] = (d > 0.f) ? rsqrtf(d) : 0.f;
  }
}

// out[i][f] = dinv[i]^2 * h[i][f]   (self-loop term; also initializes out)
__global__ void k_agg_init(float* __restrict__ out, const float* __restrict__ h,
                           const float* __restrict__ dinv, int n, int F) {
  int t = blockIdx.x * blockDim.x + threadIdx.x;
  if (t >= n * F) return;
  int i = t / F;
  float di = dinv[i];
  out[t] = di * di * h[t];
}

// out[col][f] += dinv[row]*w*dinv[col] * h[row][f]
__global__ void k_agg_scatter(float* __restrict__ out, const float* __restrict__ h,
                              const float* __restrict__ dinv,
                              const int* __restrict__ row, const int* __restrict__ col,
                              const float* __restrict__ w, int E, int F) {
  int t = blockIdx.x * blockDim.x + threadIdx.x;
  if (t >= E * F) return;
  int e = t / F;
  int f = t - e * F;
  float we = w[e];
  if (we == 0.f) return;
  int rI = row[e], c = col[e];
  float coef = dinv[rI] * we * dinv[c];
  atomicAdd(&out[c * F + f], coef * h[rI * F + f]);
}

__global__ void k_bias_relu(float* out, const float* __restrict__ b, int n, int F) {
  int t = blockIdx.x * blockDim.x + threadIdx.x;
  if (t >= n * F) return;
  float v = out[t] + b[t % F];
  out[t] = v > 0.f ? v : 0.f;
}

// ---------------- SAGPool kernels ----------------
// aggr[col][f] += mask * x[row][f]   (scorer uses NO edge weight)
__global__ void k_pool_aggr(float* __restrict__ aggr, const float* __restrict__ x,
                            const int* __restrict__ row, const int* __restrict__ col,
                            const int* __restrict__ mask, int E, int F) {
  int t = blockIdx.x * blockDim.x + threadIdx.x;
  if (t >= E * F) return;
  int e = t / F;
  if (mask && !mask[e]) return;
  int f = t - e * F;
  atomicAdd(&aggr[col[e] * F + f], x[row[e] * F + f]);
}

__global__ void k_pool_score(float* __restrict__ score, const float* __restrict__ aggr,
                             const float* __restrict__ x, const float* __restrict__ relw,
                             const float* __restrict__ relb, const float* __restrict__ rootw,
                             int n, int F) {
  int i = blockIdx.x * blockDim.x + threadIdx.x;
  if (i >= n) return;
  float s = relb[0];
  for (int f = 0; f < F; ++f) s += aggr[i * F + f] * relw[f];
  for (int f = 0; f < F; ++f) s += x[i * F + f] * rootw[f];
  score[i] = s;
}

__global__ void k_sort_init(float* skey, int* sidx, const float* __restrict__ score,
                            int n, int P) {
  int i = blockIdx.x * blockDim.x + threadIdx.x;
  if (i >= P) return;
  skey[i] = (i < n) ? score[i] : -INFINITY;
  sidx[i] = i;
}

// descending by key, ties broken by index ascending (stable argsort semantics)
__device__ __forceinline__ bool prec(float a, int ia, float b, int ib) {
  return (a > b) || (a == b && ia < ib);
}
__global__ void k_bitonic_step(float* skey, int* sidx, int P, int kdist, int j) {
  int i = blockIdx.x * blockDim.x + threadIdx.x;
  if (i >= P) return;
  int ixj = i ^ j;
  if (ixj <= i) return;
  float a = skey[i], b = skey[ixj];
  int ia = sidx[i], ib = sidx[ixj];
  bool iFirst = prec(a, ia, b, ib);
  bool up = ((i & kdist) == 0);
  bool doSwap = up ? !iFirst : iFirst;
  if (doSwap) {
    skey[i] = b; skey[ixj] = a;
    sidx[i] = ib; sidx[ixj] = ia;
  }
}

// xnew[j][f] = x[perm[j]][f] * tanh(score[perm[j]])
__global__ void k_pool_gather(float* __restrict__ xnew, const float* __restrict__ x,
                              const float* __restrict__ skey, const int* __restrict__ sidx,
                              int k, int F) {
  int t = blockIdx.x * blockDim.x + threadIdx.x;
  if (t >= k * F) return;
  int j = t / F;
  int f = t - j * F;
  xnew[t] = x[sidx[j] * F + f] * tanhf(skey[j]);
}

__global__ void k_newidx_scatter(int* newidx, const int* __restrict__ sidx, int k) {
  int j = blockIdx.x * blockDim.x + threadIdx.x;
  if (j < k) newidx[sidx[j]] = j;
}

__global__ void k_edge_remap(int* __restrict__ rowO, int* __restrict__ colO,
                             float* __restrict__ wO, int* __restrict__ maskO,
                             const int* __restrict__ rowI, const int* __restrict__ colI,
                             const float* __restrict__ wI, const int* __restrict__ maskI,
                             const int* __restrict__ newidx, int E) {
  int e = blockIdx.x * blockDim.x + threadIdx.x;
  if (e >= E) return;
  int r = rowI[e], c = colI[e];
  float w = wI[e];
  int mk = maskI ? maskI[e] : 1;
  int nr = newidx[r], nc = newidx[c];
  int valid = (mk && nr >= 0 && nc >= 0) ? 1 : 0;
  rowO[e] = valid ? nr : 0;
  colO[e] = valid ? nc : 0;
  wO[e]   = valid ? w : 0.f;
  maskO[e] = valid;
}

// ---------------- host orchestration ----------------
extern "C" void kernel_launch(void* const* d_in, const int* in_sizes, int n_in,
                              void* d_out, int out_size, void* d_ws, size_t ws_size,
                              hipStream_t stream) {
  (void)in_sizes; (void)n_in; (void)out_size; (void)ws_size;
  const int N = 100000, E = 1600000, K1 = 50000, K2 = 25000;
  const int P1 = 131072, P2 = 65536;
  const int K2pad = 25008;  // K2 rounded up to 16-row WMMA tiles

  const float* x0   = (const float*)d_in[0];
  const int*   row0 = (const int*)d_in[1];        // edge_index[0]
  const int*   col0 = ((const int*)d_in[1]) + E;  // edge_index[1]
  const float* ew0  = (const float*)d_in[2];
  const float* W1 = (const float*)d_in[3];  const float* b1 = (const float*)d_in[4];
  const float* W2 = (const float*)d_in[5];  const float* b2 = (const float*)d_in[6];
  const float* p1rw = (const float*)d_in[7]; const float* p1rb = (const float*)d_in[8];
  const float* p1ow = (const float*)d_in[9];
  const float* W3 = (const float*)d_in[10]; const float* b3 = (const float*)d_in[11];
  const float* p2rw = (const float*)d_in[12]; const float* p2rb = (const float*)d_in[13];
  const float* p2ow = (const float*)d_in[14];
  const float* W4 = (const float*)d_in[15]; const float* b4 = (const float*)d_in[16];
  const float* W5 = (const float*)d_in[17]; const float* b5 = (const float*)d_in[18];

  // workspace carve (~107 MB)
  char* wp = (char*)d_ws;
  auto carve = [&](size_t bytes) -> void* {
    void* q = (void*)wp;
    wp += (bytes + 255) & ~(size_t)255;
    return q;
  };
  float* bufA   = (float*)carve((size_t)N * 64 * 4);
  float* bufB   = (float*)carve((size_t)N * 64 * 4);
  float* bufC   = (float*)carve((size_t)N * 64 * 4);
  float* degb   = (float*)carve((size_t)N * 4);
  float* score  = (float*)carve((size_t)N * 4);
  float* skey   = (float*)carve((size_t)P1 * 4);
  int*   sidx   = (int*)  carve((size_t)P1 * 4);
  int*   newidx = (int*)  carve((size_t)N * 4);
  int*   erow   = (int*)  carve((size_t)E * 4);
  int*   ecol   = (int*)  carve((size_t)E * 4);
  float* ewf    = (float*)carve((size_t)E * 4);
  int*   emask  = (int*)  carve((size_t)E * 4);
  float* xpad   = (float*)carve((size_t)N * 4 * 4);  // x widened 3 -> 4 cols
  float* wpad   = (float*)carve(4 * 16 * 4);         // W1 padded 3x16 -> 4x16

  auto conv = [&](const float* X, int Fin, int n, int Fout,
                  const float* W, const float* b,
                  const int* row, const int* col, const float* w,
                  float* h, float* out) {
    launch_gemm(Fin, Fout, X, W, h, cdiv(n, 16), stream);
    k_fill_f32<<<cdiv(n, TPB), TPB, 0, stream>>>(degb, 1.0f, n);
    k_deg_scatter<<<cdiv(E, TPB), TPB, 0, stream>>>(degb, col, w, E);
    k_rsqrt_inplace<<<cdiv(n, TPB), TPB, 0, stream>>>(degb, n);
    k_agg_init<<<cdiv(n * Fout, TPB), TPB, 0, stream>>>(out, h, degb, n, Fout);
    k_agg_scatter<<<cdiv(E * Fout, TPB), TPB, 0, stream>>>(out, h, degb, row, col, w, E, Fout);
    k_bias_relu<<<cdiv(n * Fout, TPB), TPB, 0, stream>>>(out, b, n, Fout);
  };

  auto pool = [&](const float* X, int n, int F, int k, int P,
                  const float* relw, const float* relb, const float* rootw,
                  const int* rowI, const int* colI, const float* wI, const int* maskI,
                  float* aggr, float* xnew,
                  int* rowO, int* colO, float* wO, int* maskO) {
    k_fill_f32<<<cdiv(n * F, TPB), TPB, 0, stream>>>(aggr, 0.f, n * F);
    k_pool_aggr<<<cdiv(E * F, TPB), TPB, 0, stream>>>(aggr, X, rowI, colI, maskI, E, F);
    k_pool_score<<<cdiv(n, TPB), TPB, 0, stream>>>(score, aggr, X, relw, relb, rootw, n, F);
    k_sort_init<<<cdiv(P, TPB), TPB, 0, stream>>>(skey, sidx, score, n, P);
    for (int kk = 2; kk <= P; kk <<= 1)
      for (int j = kk >> 1; j > 0; j >>= 1)
        k_bitonic_step<<<cdiv(P, TPB), TPB, 0, stream>>>(skey, sidx, P, kk, j);
    k_pool_gather<<<cdiv(k * F, TPB), TPB, 0, stream>>>(xnew, X, skey, sidx, k, F);
    k_fill_i32<<<cdiv(n, TPB), TPB, 0, stream>>>(newidx, -1, n);
    k_newidx_scatter<<<cdiv(k, TPB), TPB, 0, stream>>>(newidx, sidx, k);
    k_edge_remap<<<cdiv(E, TPB), TPB, 0, stream>>>(rowO, colO, wO, maskO,
                                                   rowI, colI, wI, maskI, newidx, E);
  };

  // prep: widen x to [N,4], pad W1 to [4,16] so all GEMMs are guard-free
  k_pad_cols<<<cdiv(N * 4, TPB), TPB, 0, stream>>>(xpad, x0, N, 3, 4);
  k_copy_pad<<<1, 64, 0, stream>>>(wpad, W1, 48, 64);

  // conv1: [N,3->4] -> [N,16]
  conv(xpad, 4, N, 16, wpad, b1, row0, col0, ew0, bufA, bufB);
  // conv2: [N,16] -> [N,32]
  conv(bufB, 16, N, 32, W2, b2, row0, col0, ew0, bufA, bufC);
  // pool1: keep K1, x3 = bufB [K1,32]; edges -> ws arrays
  pool(bufC, N, 32, K1, P1, p1rw, p1rb, p1ow,
       row0, col0, ew0, nullptr, bufA, bufB, erow, ecol, ewf, emask);
  // conv3: [K1,32] -> [K1,64]   (K1 is a multiple of 16)
  conv(bufB, 32, K1, 64, W3, b3, erow, ecol, ewf, bufA, bufC);
  // pool2: keep K2, x4 = bufB [K2,64]; edges remapped in place (elementwise)
  pool(bufC, K1, 64, K2, P2, p2rw, p2rb, p2ow,
       erow, ecol, ewf, emask, bufA, bufB, erow, ecol, ewf, emask);
  // zero pad rows K2..K2pad of the GEMM input, then conv4: [K2,64] -> [K2,32]
  k_fill_f32<<<cdiv((K2pad - K2) * 64, TPB), TPB, 0, stream>>>(bufB + (size_t)K2 * 64, 0.f,
                                                               (K2pad - K2) * 64);
  conv(bufB, 64, K2, 32, W4, b4, erow, ecol, ewf, bufA, bufC);
  // zero pad rows, then conv5: [K2,32] -> [K2,32], aggregated into d_out
  k_fill_f32<<<cdiv((K2pad - K2) * 32, TPB), TPB, 0, stream>>>(bufC + (size_t)K2 * 32, 0.f,
                                                               (K2pad - K2) * 32);
  conv(bufC, 32, K2, 32, W5, b5, erow, ecol, ewf, bufA, (float*)d_out);
}